// VmLoss_50285477101996
// MI455X (gfx1250) — compile-verified
//
#include <hip/hip_runtime.h>
#include <hip/hip_bf16.h>
#include <math.h>

typedef float v2f __attribute__((ext_vector_type(2)));
typedef float v8f __attribute__((ext_vector_type(8)));

// ---------------------------------------------------------------------------
// Kernel 1: per pose pair b, build dRT = RT(target) - RT(output)  (3x4), and
// scatter into the WMMA A-matrix staging buffer: one 16x4 row-major block per
// group of 4 poses. Row mapping inside a block:
//   pose q=0 -> rows 0..2, q=1 -> rows 3..5, q=2 -> rows 8..10, q=3 -> rows 11..13
//   rows 6,7,14,15 are zero padding.
// This matches the 32-bit C/D VGPR layout (rows 0-7 -> lanes 0-15, rows 8-15
// -> lanes 16-31) so each lane later owns a full 3-vector displacement.
// ---------------------------------------------------------------------------
__device__ inline void pose_to_rt(const float* p6, const float rc[3], float RT[3][4]) {
    const float d2r = 0.017453292519943295f;
    float tx = p6[0] * d2r, ty = p6[1] * d2r, tz = p6[2] * d2r;
    float cx = cosf(tx), sx = sinf(tx);
    float cy = cosf(ty), sy = sinf(ty);
    float cz = cosf(tz), sz = sinf(tz);
    float Rz[3][3] = {{cz, sz, 0.f}, {-sz, cz, 0.f}, {0.f, 0.f, 1.f}};
    float Ry[3][3] = {{cy, 0.f, -sy}, {0.f, 1.f, 0.f}, {sy, 0.f, cy}};
    float Rx[3][3] = {{1.f, 0.f, 0.f}, {0.f, cx, sx}, {0.f, -sx, cx}};
    float T[3][3], M[3][3];
#pragma unroll
    for (int i = 0; i < 3; ++i)
#pragma unroll
        for (int j = 0; j < 3; ++j) {
            float s = 0.f;
#pragma unroll
            for (int k = 0; k < 3; ++k) s += Rz[i][k] * Ry[k][j];
            T[i][j] = s;
        }
#pragma unroll
    for (int i = 0; i < 3; ++i)
#pragma unroll
        for (int j = 0; j < 3; ++j) {
            float s = 0.f;
#pragma unroll
            for (int k = 0; k < 3; ++k) s += T[i][k] * Rx[k][j];
            M[i][j] = s;
        }
#pragma unroll
    for (int i = 0; i < 3; ++i) {
        RT[i][0] = M[i][0]; RT[i][1] = M[i][1]; RT[i][2] = M[i][2];
        RT[i][3] = rc[i] + p6[3 + i]
                 - (M[i][0] * rc[0] + M[i][1] * rc[1] + M[i][2] * rc[2]);
    }
}

__global__ void vm_prep(const float* __restrict__ outp,
                        const float* __restrict__ targ,
                        const float* __restrict__ rot_cen,
                        float* __restrict__ Amat, int B) {
    int b = blockIdx.x * blockDim.x + threadIdx.x;
    if (b >= B) return;
    float rc[3] = {rot_cen[0], rot_cen[1], rot_cen[2]};
    float R1[3][4], R2[3][4];
    pose_to_rt(targ + b * 6, rc, R1);   // rt1 (target)
    pose_to_rt(outp + b * 6, rc, R2);   // rt2 (output)
    int g = b >> 2, q = b & 3;
    int rowbase = (q < 2) ? 3 * q : 3 * q + 2;
    float* base = Amat + g * 64;        // 16 rows x 4 cols per group
#pragma unroll
    for (int i = 0; i < 3; ++i)
#pragma unroll
        for (int j = 0; j < 4; ++j)
            base[(rowbase + i) * 4 + j] = R1[i][j] - R2[i][j];
    if (q == 0) {
#pragma unroll
        for (int j = 0; j < 4; ++j) {
            base[6 * 4 + j]  = 0.f; base[7 * 4 + j]  = 0.f;
            base[14 * 4 + j] = 0.f; base[15 * 4 + j] = 0.f;
        }
    }
}

// ---------------------------------------------------------------------------
// Kernel 2: one wave per 16-point cube tile. B operand (4x16) holds the tile's
// homogeneous coords:
//   lanes 0-15 : VGPR0 = x (K=0), VGPR1 = y (K=1)
//   lanes 16-31: VGPR0 = z (K=2), VGPR1 = 1 (K=3)
// Loop over 128 pose groups; each V_WMMA_F32_16X16X4_F32 produces, per lane,
// the 3 displacement components of two poses (accumulator VGPRs 0-2 and 3-5).
// Hardware v_sqrt_f32 (via __builtin_amdgcn_sqrtf) keeps the per-iteration
// VALU tail short instead of the IEEE correction sequence.
// ---------------------------------------------------------------------------
__global__ void __launch_bounds__(256)
vm_main(const float* __restrict__ cube,
        const float* __restrict__ Amat,
        float* __restrict__ partials,
        int num_groups) {
    const int lane = threadIdx.x & 31;
    const int waveInBlock = threadIdx.x >> 5;
    const int tile = blockIdx.x * (blockDim.x >> 5) + waveInBlock;
    const int l16 = lane & 15;
    const bool hi = lane >= 16;

    const int pt = tile * 16 + l16;
    v2f bvec;
    if (!hi) { bvec.x = cube[4 * pt + 0]; bvec.y = cube[4 * pt + 1]; }
    else     { bvec.x = cube[4 * pt + 2]; bvec.y = 1.0f; }
    const float w = cube[4 * pt + 3];

    const int arow_off = l16 * 4 + (hi ? 2 : 0);

    float acc = 0.f;
#pragma unroll 4
    for (int g = 0; g < num_groups; ++g) {
        const float* ab = Amat + g * 64 + arow_off;
        v2f avec; avec.x = ab[0]; avec.y = ab[1];
        v8f c = {};
        c = __builtin_amdgcn_wmma_f32_16x16x4_f32(
                /*neg_a=*/false, avec, /*neg_b=*/false, bvec,
                /*c_mod=*/(short)0, c, /*reuse_a=*/false, /*reuse_b=*/false);
        float n0 = c[0] * c[0] + c[1] * c[1] + c[2] * c[2];
        float n1 = c[3] * c[3] + c[4] * c[4] + c[5] * c[5];
        float d0 = __builtin_amdgcn_sqrtf(n0);   // single v_sqrt_f32
        float d1 = __builtin_amdgcn_sqrtf(n1);
        acc += w * (d0 + d1);
    }

    // wave32 reduction
#pragma unroll
    for (int off = 16; off > 0; off >>= 1)
        acc += __shfl_xor(acc, off, 32);
    if (lane == 0) partials[tile] = acc;
}

// ---------------------------------------------------------------------------
// Kernel 3: deterministic tree reduction of per-tile partials, scale by
// 1/(B*N).
// ---------------------------------------------------------------------------
__global__ void vm_reduce(const float* __restrict__ partials, float* __restrict__ out,
                          int ntiles, float scale) {
    __shared__ float sm[256];
    float s = 0.f;
    for (int i = threadIdx.x; i < ntiles; i += 256) s += partials[i];
    sm[threadIdx.x] = s;
    __syncthreads();
    for (int stride = 128; stride > 0; stride >>= 1) {
        if ((int)threadIdx.x < stride) sm[threadIdx.x] += sm[threadIdx.x + stride];
        __syncthreads();
    }
    if (threadIdx.x == 0) out[0] = sm[0] * scale;
}

extern "C" void kernel_launch(void* const* d_in, const int* in_sizes, int n_in,
                              void* d_out, int out_size, void* d_ws, size_t ws_size,
                              hipStream_t stream) {
    const float* outp    = (const float*)d_in[0];  // (B,6)
    const float* targ    = (const float*)d_in[1];  // (B,6)
    const float* cube    = (const float*)d_in[2];  // (N,4)
    const float* rot_cen = (const float*)d_in[3];  // (3,)
    float* out = (float*)d_out;

    const int B  = in_sizes[0] / 6;   // 512
    const int N  = in_sizes[2] / 4;   // 32768
    const int num_groups = B / 4;     // 128 pose groups (4 poses per WMMA)
    const int ntiles = N / 16;        // 2048 point tiles (one wave each)

    float* Amat     = (float*)d_ws;                                 // num_groups*64 floats
    float* partials = (float*)((char*)d_ws + (size_t)num_groups * 64 * sizeof(float));

    // Kernel 1: build packed dRT A-matrices
    vm_prep<<<(B + 255) / 256, 256, 0, stream>>>(outp, targ, rot_cen, Amat, B);

    // Kernel 2: WMMA main loop — 8 waves (one tile each) per 256-thread block
    const int wavesPerBlock = 8;
    vm_main<<<ntiles / wavesPerBlock, wavesPerBlock * 32, 0, stream>>>(
        cube, Amat, partials, num_groups);

    // Kernel 3: final reduction
    const float scale = 1.0f / ((float)B * (float)N);
    vm_reduce<<<1, 256, 0, stream>>>(partials, out, ntiles, scale);
}